// AggrOp_10496900072252
// MI455X (gfx1250) — compile-verified
//
#include <hip/hip_runtime.h>

typedef __attribute__((ext_vector_type(2))) float v2f;
typedef __attribute__((ext_vector_type(8))) float v8f;

#define IN_DIM  128
#define OUT_DIM 128

// ---------------------------------------------------------------------------
// Kernel 0: Wsum = W0 + W1 + W2   (16384 elements, trivial)
// ---------------------------------------------------------------------------
__global__ void wsum_kernel(const float* __restrict__ W0,
                            const float* __restrict__ W1,
                            const float* __restrict__ W2,
                            float* __restrict__ Wsum) {
  int i = blockIdx.x * blockDim.x + threadIdx.x;
  if (i < IN_DIM * OUT_DIM) Wsum[i] = W0[i] + W1[i] + W2[i];
}

// ---------------------------------------------------------------------------
// Kernel 1: H = x @ Wsum + one_hot_h, fp32 WMMA (V_WMMA_F32_16X16X4_F32).
// Block = 128 threads = 4 waves; each wave owns a 16-row x 128-col slab.
// W (64 KB) staged in LDS once per block (conflict-free row-major layout).
// A fragment for iteration k+1 is prefetched before issuing iteration k's
// WMMA block so the global-load wait overlaps 8 WMMAs + 8 DS loads.
//
// Fragment layouts per CDNA5 ISA (32-bit, wave32):
//   A 16x4:  lanes 0-15 -> row M=lane, v0=K0 v1=K1; lanes 16-31 -> v0=K2 v1=K3
//   B 4x16:  lanes 0-15 -> col N=lane, v0=K0 v1=K1; lanes 16-31 -> v0=K2 v1=K3
//   C/D 16x16: VGPR i: lanes 0-15 -> (M=i, N=lane); lanes 16-31 -> (M=i+8)
// ---------------------------------------------------------------------------
__global__ void gemm_wmma_kernel(const float* __restrict__ x,
                                 const float* __restrict__ oneh,
                                 const float* __restrict__ Wsum,
                                 float* __restrict__ H, int N) {
  __shared__ float Wl[IN_DIM * OUT_DIM];   // 64 KB (WGP has 320 KB LDS)
  {
    const float4* src = (const float4*)Wsum;
    float4* dst = (float4*)Wl;
    for (int i = threadIdx.x; i < IN_DIM * OUT_DIM / 4; i += blockDim.x)
      dst[i] = src[i];
  }
  __syncthreads();

  const int lane = threadIdx.x & 31;   // wave32 (gfx1250)
  const int wave = threadIdx.x >> 5;
  const int base = (blockIdx.x * 4 + wave) * 16;
  if (base >= N) return;               // whole-wave exit; no later barriers

  const int half = lane >> 4;          // 0 = lanes 0-15, 1 = lanes 16-31
  const int l16  = lane & 15;

  // Init accumulators from one_hot_h (folds the "+ one_hot_h" into C).
  v8f acc[8];
#pragma unroll
  for (int nt = 0; nt < 8; ++nt) {
#pragma unroll
    for (int i = 0; i < 8; ++i) {
      int r  = base + i + half * 8;
      int rc = r < N ? r : N - 1;      // clamp; garbage rows masked at store
      acc[nt][i] = oneh[(size_t)rc * OUT_DIM + nt * 16 + l16];
    }
  }

  int arow = base + l16;
  if (arow >= N) arow = N - 1;
  const float* aptr = x + (size_t)arow * IN_DIM + half * 2;

  v2f a_cur;
  a_cur[0] = aptr[0];
  a_cur[1] = aptr[1];

  for (int k0 = 0; k0 < IN_DIM; k0 += 4) {
    // Prefetch next A fragment (uniform branch; no OOB past the row end).
    v2f a_nxt = a_cur;
    if (k0 + 4 < IN_DIM) {
      a_nxt[0] = aptr[k0 + 4];
      a_nxt[1] = aptr[k0 + 5];
    }
#pragma unroll
    for (int nt = 0; nt < 8; ++nt) {
      v2f b;
      b[0] = Wl[(k0 + half * 2    ) * OUT_DIM + nt * 16 + l16];
      b[1] = Wl[(k0 + half * 2 + 1) * OUT_DIM + nt * 16 + l16];
      acc[nt] = __builtin_amdgcn_wmma_f32_16x16x4_f32(
          /*neg_a=*/false, a_cur, /*neg_b=*/false, b,
          /*c_mod=*/(short)0, acc[nt], /*reuse_a=*/false, /*reuse_b=*/false);
    }
    a_cur = a_nxt;
  }

#pragma unroll
  for (int nt = 0; nt < 8; ++nt) {
#pragma unroll
    for (int i = 0; i < 8; ++i) {
      int r = base + i + half * 8;
      if (r < N) H[(size_t)r * OUT_DIM + nt * 16 + l16] = acc[nt][i];
    }
  }
}

// ---------------------------------------------------------------------------
// Kernel 2: out[rows[e]] += vals[e] * H[cols[e]], one wave per edge.
// 32 lanes x float4 = one 512 B row per wave (coalesced); scatter via
// unsafeAtomicAdd -> hardware global_atomic_add_f32 (no-return, no CAS loop),
// relaxed device scope; both H and out are L2-resident (192 MB L2).
// ---------------------------------------------------------------------------
__global__ void edge_scatter_kernel(const float* __restrict__ H,
                                    const int* __restrict__ rows,
                                    const int* __restrict__ cols,
                                    const float* __restrict__ vals,
                                    float* __restrict__ out, int nE) {
  const int lane  = threadIdx.x & 31;
  int       warp  = (blockIdx.x * blockDim.x + threadIdx.x) >> 5;
  const int nwarp = (gridDim.x * blockDim.x) >> 5;

  for (int e = warp; e < nE; e += nwarp) {
    const int   r = rows[e];
    const int   c = cols[e];
    const float v = vals[e];
    float4 h = ((const float4*)(H + (size_t)c * OUT_DIM))[lane];
    float* dst = out + (size_t)r * OUT_DIM + lane * 4;
    unsafeAtomicAdd(dst + 0, v * h.x);
    unsafeAtomicAdd(dst + 1, v * h.y);
    unsafeAtomicAdd(dst + 2, v * h.z);
    unsafeAtomicAdd(dst + 3, v * h.w);
  }
}

// ---------------------------------------------------------------------------
extern "C" void kernel_launch(void* const* d_in, const int* in_sizes, int n_in,
                              void* d_out, int out_size, void* d_ws, size_t ws_size,
                              hipStream_t stream) {
  const float* x    = (const float*)d_in[0];
  const float* oneh = (const float*)d_in[1];
  const float* W0   = (const float*)d_in[2];
  const float* W1   = (const float*)d_in[3];
  const float* W2   = (const float*)d_in[4];
  const int*   rows = (const int*)d_in[5];
  const int*   cols = (const int*)d_in[6];
  const float* vals = (const float*)d_in[7];

  const int N = in_sizes[0] / IN_DIM;    // 100000
  const int E = in_sizes[5];             // 1600000

  // Workspace: [Wsum: 128*128 f32][H: N*128 f32]
  float* Wsum = (float*)d_ws;
  float* H    = (float*)((char*)d_ws + (size_t)IN_DIM * OUT_DIM * sizeof(float));

  hipMemsetAsync(d_out, 0, (size_t)out_size * sizeof(float), stream);

  wsum_kernel<<<(IN_DIM * OUT_DIM + 255) / 256, 256, 0, stream>>>(W0, W1, W2, Wsum);

  const int gblocks = (N + 63) / 64;     // 4 waves * 16 rows per block
  gemm_wmma_kernel<<<gblocks, 128, 0, stream>>>(x, oneh, Wsum, H, N);

  edge_scatter_kernel<<<8192, 256, 0, stream>>>(H, rows, cols, vals, (float*)d_out, E);
}